// Loss_29789893165394
// MI455X (gfx1250) — compile-verified
//
#include <hip/hip_runtime.h>
#include <hip/hip_bf16.h>
#include <math.h>

#define INSTANCE_ID 1

typedef __attribute__((ext_vector_type(4))) float f32x4;
typedef __attribute__((ext_vector_type(4))) int   i32x4;
typedef __attribute__((ext_vector_type(2))) float v2f;
typedef __attribute__((ext_vector_type(8))) float v8f;

// Full 32-lane wave sum via V_WMMA_F32_16X16X4_F32.
// A (16x4): lane L<16 supplies A[L][0] (vgpr0), A[L][1] (vgpr1);
//           lane L>=16 supplies A[L-16][2], A[L-16][3].
// With vgpr0 = x, vgpr1 = 0 and B = all-ones (layout independent):
//   D[m][n] = x_m + x_{m+16}  for every column n.
// Lane L<16 holds D[0..7][L] in d0..d7; lane L>=16 holds D[8..15][L-16].
// Sum the 8 D regs, then add across wave halves -> every lane has the total.
__device__ __forceinline__ float wave_sum32(float x) {
    v2f a; a[0] = x;    a[1] = 0.0f;
    v2f b; b[0] = 1.0f; b[1] = 1.0f;
    v8f c = {};
    c = __builtin_amdgcn_wmma_f32_16x16x4_f32(false, a, false, b,
                                              (short)0, c, false, false);
    float s = ((c[0] + c[1]) + (c[2] + c[3])) + ((c[4] + c[5]) + (c[6] + c[7]));
    s += __shfl_xor(s, 16, 32);
    return s;
}

// Per-ray contribution: 3-channel squared diffs + masked obj term + opacity term.
#define RAYTERM(ax,ay,az, sx,sy,sz, ox,oy,oz, opv, pidv) do {              \
    float d0 = (ax) - (sx), d1 = (ay) - (sy), d2 = (az) - (sz);            \
    float ds = d0*d0 + d1*d1 + d2*d2;                                      \
    float e0 = (ax) - (ox), e1 = (ay) - (oy), e2 = (az) - (oz);            \
    float dobj = e0*e0 + e1*e1 + e2*e2;                                    \
    float m = (inst[(pidv)] == INSTANCE_ID) ? 1.0f : 0.0f;                 \
    acc_s += ds;                                                           \
    acc_o += m * dobj;                                                     \
    float em = m - (opv);                                                  \
    acc_p += em * em;                                                      \
} while (0)

__global__ void __launch_bounds__(256)
loss_partials(const float* __restrict__ rays,
              const float* __restrict__ scn,
              const float* __restrict__ obj,
              const float* __restrict__ opac,
              const int*   __restrict__ pix,
              const int*   __restrict__ inst,
              float*       __restrict__ partials,   // [3 * gridDim.x]
              int n_rays) {
    const int  tid     = threadIdx.x;
    const long gid     = (long)blockIdx.x * blockDim.x + tid;
    const long stride4 = (long)gridDim.x * blockDim.x * 4;

    float acc_s = 0.0f, acc_o = 0.0f, acc_p = 0.0f;

    for (long r = gid * 4; r < (long)n_rays; r += stride4) {
        const f32x4* pr = (const f32x4*)(rays + r * 3);
        const f32x4* ps = (const f32x4*)(scn  + r * 3);
        const f32x4* po = (const f32x4*)(obj  + r * 3);
        // Streamed once: non-temporal (th:TH_LOAD_NT) keeps L2 free for the
        // instance_ids gather table (4 MB, L2-resident at 192 MB).
        f32x4 a0 = __builtin_nontemporal_load(pr + 0);
        f32x4 a1 = __builtin_nontemporal_load(pr + 1);
        f32x4 a2 = __builtin_nontemporal_load(pr + 2);
        f32x4 s0 = __builtin_nontemporal_load(ps + 0);
        f32x4 s1 = __builtin_nontemporal_load(ps + 1);
        f32x4 s2 = __builtin_nontemporal_load(ps + 2);
        f32x4 o0 = __builtin_nontemporal_load(po + 0);
        f32x4 o1 = __builtin_nontemporal_load(po + 1);
        f32x4 o2 = __builtin_nontemporal_load(po + 2);
        f32x4 op = __builtin_nontemporal_load((const f32x4*)(opac + r));
        i32x4 pd = __builtin_nontemporal_load((const i32x4*)(pix  + r));

        RAYTERM(a0[0],a0[1],a0[2], s0[0],s0[1],s0[2], o0[0],o0[1],o0[2], op[0], pd[0]);
        RAYTERM(a0[3],a1[0],a1[1], s0[3],s1[0],s1[1], o0[3],o1[0],o1[1], op[1], pd[1]);
        RAYTERM(a1[2],a1[3],a2[0], s1[2],s1[3],s2[0], o1[2],o1[3],o2[0], op[2], pd[2]);
        RAYTERM(a2[1],a2[2],a2[3], s2[1],s2[2],s2[3], o2[1],o2[2],o2[3], op[3], pd[3]);
    }

    // Wave-level reduction via WMMA (EXEC is all-ones here: uniform control flow).
    acc_s = wave_sum32(acc_s);
    acc_o = wave_sum32(acc_o);
    acc_p = wave_sum32(acc_p);

    __shared__ float ls[8], lo[8], lp[8];
    const int wave = tid >> 5;
    if ((tid & 31) == 0) { ls[wave] = acc_s; lo[wave] = acc_o; lp[wave] = acc_p; }
    __syncthreads();
    if (tid == 0) {
        float S = 0.0f, O = 0.0f, P = 0.0f;
        #pragma unroll
        for (int w = 0; w < 8; ++w) { S += ls[w]; O += lo[w]; P += lp[w]; }
        const int nb = gridDim.x;
        partials[blockIdx.x]          = S;
        partials[nb + blockIdx.x]     = O;
        partials[2 * nb + blockIdx.x] = P;
    }
}

__global__ void __launch_bounds__(256)
loss_finalize(const float* __restrict__ partials, int nblocks,
              float* __restrict__ out, float inv_n) {
    const int tid = threadIdx.x;
    float S = 0.0f, O = 0.0f, P = 0.0f;
    for (int i = tid; i < nblocks; i += 256) {
        S += partials[i];
        O += partials[nblocks + i];
        P += partials[2 * nblocks + i];
    }
    S = wave_sum32(S);
    O = wave_sum32(O);
    P = wave_sum32(P);

    __shared__ float ls[8], lo[8], lp[8];
    const int wave = tid >> 5;
    if ((tid & 31) == 0) { ls[wave] = S; lo[wave] = O; lp[wave] = P; }
    __syncthreads();
    if (tid == 0) {
        float Ss = 0.0f, Oo = 0.0f, Pp = 0.0f;
        #pragma unroll
        for (int w = 0; w < 8; ++w) { Ss += ls[w]; Oo += lo[w]; Pp += lp[w]; }
        float mean_scn     = Ss * inv_n;
        float mean_obj     = Oo * inv_n;
        float color_loss   = mean_scn + mean_obj;     // W_COLOR = W_SCN = 1
        float opacity_loss = Pp * inv_n;              // W_OPACITY = 1
        float loss         = color_loss + opacity_loss;
        float psnr_scn = -10.0f * log10f(mean_scn);
        float psnr_obj = -10.0f * log10f(mean_obj);
        if (isinf(psnr_scn)) psnr_scn = 0.0f;
        if (isinf(psnr_obj)) psnr_obj = 0.0f;
        out[0] = loss;
        out[1] = color_loss;
        out[2] = opacity_loss;
        out[3] = psnr_scn;
        out[4] = psnr_obj;
    }
}

extern "C" void kernel_launch(void* const* d_in, const int* in_sizes, int n_in,
                              void* d_out, int out_size, void* d_ws, size_t ws_size,
                              hipStream_t stream) {
    (void)n_in; (void)out_size; (void)ws_size;
    const float* rays = (const float*)d_in[0];   // [1, N_RAYS, 3] f32
    const float* scn  = (const float*)d_in[1];   // [1, N_RAYS, 3] f32
    const float* obj  = (const float*)d_in[2];   // [1, N_RAYS, 3] f32
    const float* opac = (const float*)d_in[3];   // [1, N_RAYS]    f32
    const int*   pix  = (const int*)  d_in[4];   // [1, N_RAYS]    i32
    const int*   inst = (const int*)  d_in[5];   // [1, N_PIXELS]  i32
    float* out      = (float*)d_out;             // 5 scalars
    float* partials = (float*)d_ws;              // 3 * BLOCKS floats (24 KB)

    const int n_rays = in_sizes[3];              // opacity element count = N_RAYS
    const int BLOCKS = 2048;                     // 2 grid-stride iters @ 4 rays/thread

    loss_partials<<<BLOCKS, 256, 0, stream>>>(rays, scn, obj, opac, pix, inst,
                                              partials, n_rays);
    loss_finalize<<<1, 256, 0, stream>>>(partials, BLOCKS, out,
                                         1.0f / (float)n_rays);
}